// ConvS5SSM_79542794322242
// MI455X (gfx1250) — compile-verified
//
#include <hip/hip_runtime.h>
#include <hip/hip_bf16.h>

typedef __attribute__((ext_vector_type(2))) float v2f;
typedef __attribute__((ext_vector_type(8))) float v8f;

#define LSEQ 16
#define BSZ  8
#define HH   32
#define WW   32
#define UU   128
#define PP   256
#define NIMG (LSEQ*BSZ)          // 128 images
#define NPIX (NIMG*HH*WW)        // 131072 GEMM rows
#define EPSN 1e-5f

// ---------------- prep kernels ----------------

__global__ void prep_abar(const float* __restrict__ lam_re,
                          const float* __restrict__ lam_im,
                          const float* __restrict__ log_step,
                          float* __restrict__ abar /* (P,2) */) {
    int p = blockIdx.x * blockDim.x + threadIdx.x;
    if (p >= PP) return;
    float lr = fminf(lam_re[p], -1e-4f);
    float li = lam_im[p];
    float st = __expf(log_step[p]);
    abar[2*p + 0] = lr * st;
    abar[2*p + 1] = li * st;
}

// W_B[k][n], k = (ky*3+kx)*128 + cin (K=1152), n<256: real P=n, n>=256: imag P=n-256
__global__ void prep_WB(const float* __restrict__ B_ri,   // (P,U,3,3,2)
                        const float* __restrict__ log_step,
                        float* __restrict__ WB) {
    int t = blockIdx.x * blockDim.x + threadIdx.x;
    const int TOT = 9*UU*512;
    if (t >= TOT) return;
    int k = t >> 9;            // /512
    int n = t & 511;
    int key = k / UU;          // 0..8
    int cin = k - key*UU;
    int ky = key / 3, kx = key - 3*ky;
    int p  = n & 255;
    int ri = n >> 8;
    float st = __expf(log_step[p]);
    int bidx = ((((p*UU + cin)*3 + ky)*3 + kx) << 1) + ri;
    WB[t] = B_ri[bidx] * st;
}

// W_C[k][n], k = (ky*3+kx)*512 + c (K=4608), c<256 real(+2*Cr), else imag(-2*Ci); n = out channel
__global__ void prep_WC(const float* __restrict__ C_ri,   // (U,P,3,3,2)
                        float* __restrict__ WC) {
    int t = blockIdx.x * blockDim.x + threadIdx.x;
    const int TOT = 9*512*UU;
    if (t >= TOT) return;
    int k  = t >> 7;           // /128
    int uo = t & 127;
    int key = k >> 9;          // /512 -> 0..8
    int c   = k & 511;
    int ky = key / 3, kx = key - 3*ky;
    int p  = c & 255;
    int ri = c >> 8;
    float sgn = (ri == 0) ? 2.0f : -2.0f;
    int cidx = ((((uo*PP + p)*3 + ky)*3 + kx) << 1) + ri;
    WC[t] = sgn * C_ri[cidx];
}

// ---------------- implicit-GEMM conv (3x3 SAME) via V_WMMA_F32_16X16X4_F32 ----------------
// Double-buffered LDS, register-staged prefetch, one barrier per K-stage.
// in : (NIMG, 32, 32, CIN) fp32   Wk : (9*CIN, COUT)   out : (NPIX, COUT)
// Block: 256 threads = 8 waves (4x2), block tile 128(M) x 64(N), K-stage 16.
template<int CIN, int COUT>
__global__ __launch_bounds__(256)
void conv_gemm_wmma(const float* __restrict__ in,
                    const float* __restrict__ Wk,
                    float* __restrict__ out) {
    __shared__ float sA[2][128][17];   // +1 pad: conflict-free frag reads
    __shared__ float sB[2][16][72];    // stride 72: half-wave bank sets disjoint

    const int tid  = threadIdx.x;
    const int wave = tid >> 5;
    const int lane = tid & 31;
    const int wm = wave >> 1;          // 0..3
    const int wn = wave & 1;           // 0..1
    const int lhalf = lane >> 4;       // K-pair selector (ISA 16x4 f32 A layout)
    const int l16   = lane & 15;

    const int m0 = blockIdx.x * 128;
    const int n0 = blockIdx.y * 64;
    const int K  = 9 * CIN;

    v8f acc[2][2];
    acc[0][0] = (v8f)0.0f; acc[0][1] = (v8f)0.0f;
    acc[1][0] = (v8f)0.0f; acc[1][1] = (v8f)0.0f;

    float4 aReg[2];
    float4 bReg;

    // ---- register-stage loaders for K-stage starting at k0 ----
    auto loadA = [&](int k0, float4 aR[2]) {
        const int key  = k0 / CIN;               // 16 | CIN -> stage within one tap
        const int cin0 = k0 - key * CIN;
        const int ky = key / 3 - 1;
        const int kx = key - 3*(key/3) - 1;
        #pragma unroll
        for (int i = 0; i < 2; ++i) {
            int idx = tid + i*256;               // 0..511 float4 slots (128 rows x 4)
            int row = idx >> 2;
            int kq  = idx & 3;
            int m = m0 + row;
            int nimg = m >> 10;
            int hw = m & 1023;
            int h = (hw >> 5) + ky;
            int w = (hw & 31) + kx;
            float4 v = make_float4(0.f, 0.f, 0.f, 0.f);
            if ((unsigned)h < 32u && (unsigned)w < 32u)
                v = *(const float4*)(in + (size_t)(nimg*1024 + h*32 + w) * CIN + cin0 + kq*4);
            aR[i] = v;
        }
    };
    auto loadB = [&](int k0, float4& bR) {
        int kr = tid >> 4;                       // 0..15
        int nq = tid & 15;                       // float4 col group
        bR = *(const float4*)(Wk + (size_t)(k0 + kr) * COUT + n0 + nq*4);
    };
    auto stage = [&](int b, const float4 aR[2], const float4& bR) {
        #pragma unroll
        for (int i = 0; i < 2; ++i) {
            int idx = tid + i*256;
            int row = idx >> 2;
            int kq  = idx & 3;
            sA[b][row][kq*4+0] = aR[i].x;
            sA[b][row][kq*4+1] = aR[i].y;
            sA[b][row][kq*4+2] = aR[i].z;
            sA[b][row][kq*4+3] = aR[i].w;
        }
        int kr = tid >> 4;
        int nq = tid & 15;
        sB[b][kr][nq*4+0] = bR.x;
        sB[b][kr][nq*4+1] = bR.y;
        sB[b][kr][nq*4+2] = bR.z;
        sB[b][kr][nq*4+3] = bR.w;
    };

    loadA(0, aReg);
    loadB(0, bReg);

    int buf = 0;
    for (int k0 = 0; k0 < K; k0 += 16) {
        stage(buf, aReg, bReg);
        __syncthreads();

        // prefetch next stage while WMMAs consume current LDS buffer
        if (k0 + 16 < K) {
            loadA(k0 + 16, aReg);
            loadB(k0 + 16, bReg);
        }

        #pragma unroll
        for (int kk = 0; kk < 4; ++kk) {
            const int kb = kk*4 + lhalf*2;       // lanes 0-15: K0/K1, lanes 16-31: K2/K3
            v2f af[2], bf[2];
            #pragma unroll
            for (int ms = 0; ms < 2; ++ms) {
                int row = wm*32 + ms*16 + l16;
                af[ms].x = sA[buf][row][kb];
                af[ms].y = sA[buf][row][kb+1];
            }
            #pragma unroll
            for (int ns = 0; ns < 2; ++ns) {
                int col = wn*32 + ns*16 + l16;
                bf[ns].x = sB[buf][kb][col];
                bf[ns].y = sB[buf][kb+1][col];
            }
            #pragma unroll
            for (int ms = 0; ms < 2; ++ms)
                #pragma unroll
                for (int ns = 0; ns < 2; ++ns)
                    acc[ms][ns] = __builtin_amdgcn_wmma_f32_16x16x4_f32(
                        false, af[ms], false, bf[ns], (short)0, acc[ms][ns], false, false);
        }
        buf ^= 1;
    }

    // ---- store: VGPR r -> M = base + r (lanes 0-15) / base + 8 + r (lanes 16-31), N = l16 ----
    #pragma unroll
    for (int ms = 0; ms < 2; ++ms) {
        #pragma unroll
        for (int ns = 0; ns < 2; ++ns) {
            int mbase = m0 + wm*32 + ms*16 + lhalf*8;
            int col   = n0 + wn*32 + ns*16 + l16;
            #pragma unroll
            for (int r = 0; r < 8; ++r)
                out[(size_t)(mbase + r) * COUT + col] = acc[ms][ns][r];
        }
    }
}

// ---------------- diagonal scan over L (in-place over Bus) + x_last output ----------------
__global__ void scan_seq(float* __restrict__ bus,        // (L, BSZ, 1024, 512) r|i planes
                         const float* __restrict__ x0,   // (BSZ, 1024, 256)
                         const float* __restrict__ abar, // (256, 2)
                         float* __restrict__ xlast) {    // (BSZ, 1024, 256, 2)
    int t = blockIdx.x * blockDim.x + threadIdx.x;       // BSZ*1024*256 threads
    if (t >= BSZ*1024*PP) return;
    int b  = t / (1024*PP);
    int r  = t - b*(1024*PP);
    int hw = r >> 8;
    int p  = r & 255;
    float ar = abar[2*p], ai = abar[2*p+1];
    float xr = x0[(b*1024 + hw)*PP + p];
    float xi = 0.0f;
    #pragma unroll
    for (int l = 0; l < LSEQ; ++l) {
        size_t base = ((size_t)(l*BSZ + b)*1024 + hw) * 512;
        float br = bus[base + p];
        float bi = bus[base + 256 + p];
        float nr = fmaf(ar, xr, fmaf(-ai, xi, br));
        float ni = fmaf(ar, xi, fmaf( ai, xr, bi));
        bus[base + p]       = nr;
        bus[base + 256 + p] = ni;
        xr = nr; xi = ni;
    }
    size_t o = ((size_t)(b*1024 + hw)*PP + p) * 2;
    xlast[o + 0] = xr;
    xlast[o + 1] = xi;
}

// ---------------- fused depthwise-D conv + GroupNorm + tanh-GELU (in-place on ys) ----------------
__global__ __launch_bounds__(256)
void gn_gelu(float* __restrict__ ys,            // (NIMG, 1024, 128), pre-GN, rewritten in place
             const float* __restrict__ u,       // (NIMG, 1024, 128)
             const float* __restrict__ Dk,      // (3,3,1,128)
             const float* __restrict__ gsc,
             const float* __restrict__ gbi) {
    __shared__ float red1[256];
    __shared__ float red2[256];
    const int n = blockIdx.x >> 5;   // image
    const int g = blockIdx.x & 31;   // group
    const int c0 = g * 4;
    const int tid = threadIdx.x;

    float v[16];
    float s = 0.f, s2 = 0.f;
    #pragma unroll
    for (int i = 0; i < 16; ++i) {
        int e  = tid + i*256;        // 0..4095
        int hw = e >> 2;
        int cl = e & 3;
        int c  = c0 + cl;
        int h = hw >> 5, w = hw & 31;
        float du = 0.f;
        #pragma unroll
        for (int dy = -1; dy <= 1; ++dy)
            #pragma unroll
            for (int dx = -1; dx <= 1; ++dx) {
                int hh = h + dy, ww = w + dx;
                if ((unsigned)hh < 32u && (unsigned)ww < 32u)
                    du = fmaf(u[(size_t)(n*1024 + hh*32 + ww)*UU + c],
                              Dk[((dy+1)*3 + (dx+1))*UU + c], du);
            }
        float val = ys[(size_t)(n*1024 + hw)*UU + c] + du;
        v[i] = val;
        s  += val;
        s2 += val*val;
    }
    red1[tid] = s; red2[tid] = s2;
    __syncthreads();
    for (int off = 128; off > 0; off >>= 1) {
        if (tid < off) { red1[tid] += red1[tid+off]; red2[tid] += red2[tid+off]; }
        __syncthreads();
    }
    const float inv_n = 1.0f / 4096.0f;
    float mean = red1[0] * inv_n;
    float var  = red2[0] * inv_n - mean*mean;
    float rstd = rsqrtf(var + EPSN);

    #pragma unroll
    for (int i = 0; i < 16; ++i) {
        int e  = tid + i*256;
        int hw = e >> 2;
        int cl = e & 3;
        int c  = c0 + cl;
        float x = (v[i] - mean) * rstd * gsc[c] + gbi[c];
        // tanh-approx GELU (jax.nn.gelu default)
        float x3 = x*x*x;
        float tt = tanhf(0.7978845608028654f * (x + 0.044715f*x3));
        ys[(size_t)(n*1024 + hw)*UU + c] = 0.5f * x * (1.0f + tt);
    }
}

// ---------------- launch ----------------
extern "C" void kernel_launch(void* const* d_in, const int* in_sizes, int n_in,
                              void* d_out, int out_size, void* d_ws, size_t ws_size,
                              hipStream_t stream) {
    const float* in_seq  = (const float*)d_in[0];
    const float* x0      = (const float*)d_in[1];
    const float* lam_re  = (const float*)d_in[2];
    const float* lam_im  = (const float*)d_in[3];
    const float* B_ri    = (const float*)d_in[4];
    const float* C_ri    = (const float*)d_in[5];
    const float* log_st  = (const float*)d_in[6];
    const float* Dk      = (const float*)d_in[7];
    const float* gsc     = (const float*)d_in[8];
    const float* gbi     = (const float*)d_in[9];

    float* out   = (float*)d_out;
    float* xlast = out;                        // (8,32,32,256,2) = 4194304 floats
    float* ys    = out + 4194304;              // (128,1024,128)  = 16777216 floats

    float* ws  = (float*)d_ws;
    float* WB   = ws;                          // 9*128*512  = 589824
    float* WC   = WB + 589824;                 // 9*512*128  = 589824
    float* abar = WC + 589824;                 // 512
    float* bus  = abar + 512;                  // 131072*512 = 67108864

    prep_abar<<<1, 256, 0, stream>>>(lam_re, lam_im, log_st, abar);
    prep_WB<<<(9*UU*512 + 255)/256, 256, 0, stream>>>(B_ri, log_st, WB);
    prep_WC<<<(9*512*UU + 255)/256, 256, 0, stream>>>(C_ri, WC);

    // B-conv: (131072 x 1152) x (1152 x 512) -> bus
    conv_gemm_wmma<UU, 512><<<dim3(NPIX/128, 512/64), 256, 0, stream>>>(in_seq, WB, bus);

    // sequential diagonal scan over L, in place; emits x_last_ri
    scan_seq<<<(BSZ*1024*PP + 255)/256, 256, 0, stream>>>(bus, x0, abar, xlast);

    // C-conv: (131072 x 4608) x (4608 x 128) -> ys (pre-GN, into d_out region as scratch)
    conv_gemm_wmma<512, UU><<<dim3(NPIX/128, UU/64), 256, 0, stream>>>(bus, WC, ys);

    // fused depthwise D + GroupNorm + GELU, rewrites ys in place
    gn_gelu<<<NIMG*32, 256, 0, stream>>>(ys, in_seq, Dk, gsc, gbi);
}